// STARLinear_43250320671142
// MI455X (gfx1250) — compile-verified
//
#include <hip/hip_runtime.h>

typedef float v2f __attribute__((ext_vector_type(2)));
typedef float v8f __attribute__((ext_vector_type(8)));

#define B_   8192
#define IN_  128
#define OUT_ 128
#define NS_  16
#define MAXG_ (B_/16 + NS_)   // 528 worst-case 16-row groups

// Weff is stored WMMA-B-fragment packed:
//   Wf[((s*32 + kk)*2 + kh)*128 + n] = float2( W[s][4kk+2kh+0][n]*G[..], W[s][4kk+2kh+1][n]*G[..] )
// so a B fragment for (kk, kh, n) is one 64-bit load.

// ---------------- K1: pack Weff = W * G, beff = b + gb, zero counts ----------
__global__ __launch_bounds__(256) void star_prep(
    const float* __restrict__ W, const float* __restrict__ G,
    const float* __restrict__ bi, const float* __restrict__ gb,
    float* __restrict__ Wf, float* __restrict__ beff, int* __restrict__ counts)
{
    int t = blockIdx.x * 256 + threadIdx.x;     // t = ((s*128)+k)*128 + n
    if (t < NS_ * IN_ * OUT_) {
        int io = t % (IN_ * OUT_);
        int n  = t & 127;
        int k  = (t >> 7) & 127;
        int s  = t >> 14;
        int kk = k >> 2, kh = (k >> 1) & 1, j = k & 1;
        size_t dst = ((size_t)(((s * 32 + kk) * 2 + kh) * 128 + n) << 1) + j;
        Wf[dst] = W[t] * G[io];
    }
    if (t < NS_ * OUT_) beff[t] = bi[t] + gb[t % OUT_];
    if (t < NS_) counts[t] = 0;
}

// ---------------- K2: histogram of scene ids ---------------------------------
__global__ __launch_bounds__(256) void star_count(
    const int* __restrict__ scene, int* __restrict__ counts)
{
    int i = blockIdx.x * 256 + threadIdx.x;
    if (i < B_) atomicAdd(&counts[scene[i] - 1], 1);
}

// ---------------- K3: serial scan + group descriptors (tiny) -----------------
__global__ void star_scan(
    const int* __restrict__ counts, int* __restrict__ binStart,
    int* __restrict__ cursors, int* __restrict__ perm,
    int* __restrict__ groupScene, int* __restrict__ groupCount,
    int* __restrict__ totalGroups)
{
    if (threadIdx.x != 0 || blockIdx.x != 0) return;
    int off = 0, g = 0;
    for (int s = 0; s < NS_; ++s) {
        int c = counts[s];
        binStart[s] = off;
        cursors[s]  = 0;
        int cap = (c + 15) & ~15;
        for (int p = c; p < cap; ++p) perm[off + p] = 0;   // safe padding rows
        for (int k = 0; k < (cap >> 4); ++k) {
            groupScene[g] = s;
            int rem = c - k * 16;
            groupCount[g] = rem > 16 ? 16 : rem;
            ++g;
        }
        off += cap;
    }
    totalGroups[0] = g;
}

// ---------------- K4: scatter row indices into scene bins --------------------
__global__ __launch_bounds__(256) void star_scatter(
    const int* __restrict__ scene, const int* __restrict__ binStart,
    int* __restrict__ cursors, int* __restrict__ perm)
{
    int i = blockIdx.x * 256 + threadIdx.x;
    if (i < B_) {
        int s = scene[i] - 1;
        int pos = atomicAdd(&cursors[s], 1);
        perm[binStart[s] + pos] = i;
    }
}

// ---------------- K5: grouped WMMA GEMM (f32, 16x16x4) -----------------------
// Block = 256 threads = 8 wave32; wave w owns N-tile [16w,16w+16) of one
// 16-row group. 32 x v_wmma_f32_16x16x4_f32, full f32 precision.
//
// A (16x4 f32, 2 VGPR): lane L -> M = L%16, regs hold K = 2*(L/16)+{0,1}
// B (4x16 f32, 2 VGPR): lane L -> N = L%16, regs hold K = 2*(L/16)+{0,1}
// C/D (16x16 f32, 8 VGPR): lane L -> N = L%16, reg r -> M = r + 8*(L/16)
__global__ __launch_bounds__(256) void star_gemm(
    const float* __restrict__ x,    const float* __restrict__ Wf,
    const float* __restrict__ beff, const int* __restrict__ perm,
    const int* __restrict__ groupScene, const int* __restrict__ groupCount,
    const int* __restrict__ totalGroups, float* __restrict__ out)
{
    int g = blockIdx.x;
    if (g >= totalGroups[0]) return;           // wave-uniform: EXEC stays all-1s

    int s   = groupScene[g];
    int cnt = groupCount[g];

    int lane = threadIdx.x & 31;
    int wave = threadIdx.x >> 5;               // N-tile index 0..7
    int m    = lane & 15;                      // M (for A) / N-in-tile (for B,D)
    int kh   = lane >> 4;                      // K half-select
    int n    = (wave << 4) + m;                // global output column

    int rowA = perm[(g << 4) + m];             // gathered x row for this lane's M
    const float* __restrict__ xrow = x + (size_t)rowA * IN_ + 2 * kh;
    // packed B fragments for this lane: stride per kk = 2*128 float2 = 512 floats
    const v2f* __restrict__ bptr =
        (const v2f*)Wf + ((size_t)(s * 32) * 2 + kh) * 128 + n;

    v8f acc = {};
    #pragma unroll 8
    for (int kk = 0; kk < IN_ / 4; ++kk) {
        v2f a = *(const v2f*)(xrow + kk * 4);   // x[rowA, 4kk+2kh : +2]  (b64)
        v2f b = bptr[(size_t)kk * 256];         // packed Weff fragment   (b64)
        acc = __builtin_amdgcn_wmma_f32_16x16x4_f32(
            /*neg_a=*/false, a, /*neg_b=*/false, b,
            /*c_mod=*/(short)0, acc, /*reuse_a=*/false, /*reuse_b=*/false);
    }

    float bb = beff[s * OUT_ + n];
    // this lane stores D rows 8*kh .. 8*kh+7 -> 8 consecutive perm entries
    const int* pg = perm + (g << 4) + 8 * kh;
    int4 p0 = *(const int4*)(pg);
    int4 p1 = *(const int4*)(pg + 4);
    int rows[8] = {p0.x, p0.y, p0.z, p0.w, p1.x, p1.y, p1.z, p1.w};

    if (cnt == 16) {                            // fast path: full group (common)
        #pragma unroll
        for (int r = 0; r < 8; ++r) {
            float v = acc[r] + bb;
            out[(size_t)rows[r] * OUT_ + n] = v > 0.0f ? v : 0.0f;
        }
    } else {                                    // bin-tail group: guard padding
        #pragma unroll
        for (int r = 0; r < 8; ++r) {
            int mrow = r + 8 * kh;
            if (mrow < cnt) {
                float v = acc[r] + bb;
                out[(size_t)rows[r] * OUT_ + n] = v > 0.0f ? v : 0.0f;
            }
        }
    }
}

// ---------------- host: workspace carve-up + launches ------------------------
extern "C" void kernel_launch(void* const* d_in, const int* in_sizes, int n_in,
                              void* d_out, int out_size, void* d_ws, size_t ws_size,
                              hipStream_t stream) {
    const float* x     = (const float*)d_in[0];
    const int*   scene = (const int*)  d_in[1];
    const float* W     = (const float*)d_in[2];
    const float* bi    = (const float*)d_in[3];
    const float* G     = (const float*)d_in[4];
    const float* gb    = (const float*)d_in[5];
    float* out = (float*)d_out;

    char* ws = (char*)d_ws;
    size_t o = 0;
    float* Wf        = (float*)(ws + o); o += (size_t)NS_ * IN_ * OUT_ * 4;  // 1 MB packed
    float* beff      = (float*)(ws + o); o += (size_t)NS_ * OUT_ * 4;
    int*   counts    = (int*)(ws + o);   o += 64;
    int*   cursors   = (int*)(ws + o);   o += 64;
    int*   binStart  = (int*)(ws + o);   o += 64;
    int*   totalG    = (int*)(ws + o);   o += 64;
    int*   groupScn  = (int*)(ws + o);   o += ((MAXG_ * 4 + 255) & ~255);
    int*   groupCnt  = (int*)(ws + o);   o += ((MAXG_ * 4 + 255) & ~255);
    int*   perm      = (int*)(ws + o);   o += ((B_ + NS_ * 16) * 4 + 255) & ~255;
    (void)o; (void)ws_size; (void)in_sizes; (void)n_in; (void)out_size;

    star_prep<<<(NS_ * IN_ * OUT_) / 256, 256, 0, stream>>>(W, G, bi, gb, Wf, beff, counts);
    star_count<<<B_ / 256, 256, 0, stream>>>(scene, counts);
    star_scan<<<1, 32, 0, stream>>>(counts, binStart, cursors, perm, groupScn, groupCnt, totalG);
    star_scatter<<<B_ / 256, 256, 0, stream>>>(scene, binStart, cursors, perm);
    star_gemm<<<MAXG_, 256, 0, stream>>>(x, Wf, beff, perm, groupScn, groupCnt, totalG, out);

    // second tuple element: pass scene through to the tail of d_out
    hipMemcpyAsync((char*)d_out + (size_t)B_ * OUT_ * sizeof(float),
                   d_in[1], (size_t)B_ * sizeof(int),
                   hipMemcpyDeviceToDevice, stream);
}